// BridgeGap_53231824666893
// MI455X (gfx1250) — compile-verified
//
#include <hip/hip_runtime.h>
#include <cstdint>

#define BATCH      16
#define HDIM       1024
#define WDIM       1024
#define WORDS      32                      // WDIM / 32
#define TILE_ROWS  32
#define HALO       4
#define IN_ROWS    (TILE_ROWS + 2*HALO)    // 40
#define DIL_ROWS   (TILE_ROWS + 4)         // 36
#define CHUNK_ROWS 8
#define NCHUNK     (IN_ROWS / CHUNK_ROWS)  // 5
#define NTHREADS   256
#define F4_PER_CHUNK   (CHUNK_ROWS * WDIM / 4)          // 2048
#define ISSUES_PER_THR (F4_PER_CHUNK / NTHREADS)        // 8 (= per-wave ASYNCcnt/chunk)
#define IMG_STRIDE (2 * HDIM * WDIM)       // elements per image (2 channels)
#define CH_STRIDE  (HDIM * WDIM)

// ---- CDNA5 async global->LDS copy (ASYNCcnt path), per CDNA5 ISA ----
__device__ __forceinline__ void async_b128_to_lds(uint32_t lds_byte_off,
                                                  const float* gaddr) {
  // VDST = LDS byte offset VGPR, VADDR = 64-bit global address VGPR pair.
  asm volatile("global_load_async_to_lds_b128 %0, %1, off"
               :: "v"(lds_byte_off), "v"(gaddr)
               : "memory");
}
// Async loads complete in order: waiting <= N leaves the newest N in flight.
__device__ __forceinline__ void wait_asynccnt_le8() {
  asm volatile("s_wait_asynccnt 8" ::: "memory");
}
__device__ __forceinline__ void wait_asynccnt_0() {
  asm volatile("s_wait_asynccnt 0" ::: "memory");
}

__device__ __forceinline__ uint32_t ballot32(bool p) {
#if __has_builtin(__builtin_amdgcn_ballot_w32)
  return (uint32_t)__builtin_amdgcn_ballot_w32(p);
#else
  return (uint32_t)__ballot(p);
#endif
}

// Horizontal 1x5 dilation of a bitmask word with neighbor words (pad = 0).
__device__ __forceinline__ uint32_t hdil5(uint32_t m, uint32_t l, uint32_t r) {
  return m | (m << 1) | (m << 2) | (m >> 1) | (m >> 2)
           | (l >> 30) | (l >> 31) | (r << 30) | (r << 31);
}

__global__ __launch_bounds__(NTHREADS)
void BridgeGap_close_ch0_kernel(const float* __restrict__ in,
                                float* __restrict__ out) {
  __shared__ __align__(16) float fbuf[2][CHUNK_ROWS * WDIM]; // 2 x 32 KB staging
  __shared__ uint32_t bufA[IN_ROWS][WORDS];  // mask, later dil  (5 KB)
  __shared__ uint32_t bufB[IN_ROWS][WORDS];  // hdil, later hero (5 KB)

  const int tid  = threadIdx.x;
  const int lane = tid & 31;
  const int img  = blockIdx.x / (HDIM / TILE_ROWS);
  const int band = blockIdx.x % (HDIM / TILE_ROWS);
  const int row0 = band * TILE_ROWS;

  const float* src = in + (size_t)img * IMG_STRIDE;   // channel 0 of image

  // Issue one chunk's async copies into staging buffer `buf`.
  auto issue_chunk = [&](int c, int buf) {
    for (int it = 0; it < ISSUES_PER_THR; ++it) {
      int f4  = it * NTHREADS + tid;               // 0..2047
      int lrc = f4 / (WDIM / 4);                   // row within chunk
      int gr  = row0 - HALO + c * CHUNK_ROWS + lrc;
      int grc = gr < 0 ? 0 : (gr >= HDIM ? HDIM - 1 : gr);   // clamp (safe addr)
      const float* gp = src + (size_t)grc * WDIM + (f4 % (WDIM / 4)) * 4;
      uint32_t loff = (uint32_t)(size_t)(&fbuf[buf][f4 * 4]);  // LDS byte offset
      async_b128_to_lds(loff, gp);
    }
  };

  // ---- Stage 1: double-buffered async staging + ballot binarization ----
  issue_chunk(0, 0);
  for (int c = 0; c < NCHUNK; ++c) {
    const bool more = (c + 1 < NCHUNK);
    if (more) issue_chunk(c + 1, (c + 1) & 1);     // overlap next chunk's DMA
    if (more) wait_asynccnt_le8();                 // chunk c's 8 issues complete
    else      wait_asynccnt_0();
    __syncthreads();                               // chunk c visible block-wide
    const float* fb = fbuf[c & 1];
    for (int it = 0; it < (CHUNK_ROWS * WDIM) / NTHREADS; ++it) {
      int p  = it * NTHREADS + tid;                // 0..8191
      int lr = c * CHUNK_ROWS + p / WDIM;          // local mask row
      int gr = row0 - HALO + lr;
      float v = fb[p];
      bool pred = (gr >= 0) && (gr < HDIM) && ((1.0f - v) > (1.0f / 3.0f));
      uint32_t w = ballot32(pred);
      if (lane == 0) bufA[lr][(p % WDIM) >> 5] = w;
    }
    __syncthreads();             // reads of fbuf[c&1] done before chunk c+2 reuse
  }

  // ---- Stage 2a: horizontal dilation (pad 0) : bufA(mask) -> bufB(hdil) ----
  for (int i = tid; i < IN_ROWS * WORDS; i += NTHREADS) {
    int lr = i / WORDS, w = i % WORDS;
    uint32_t m = bufA[lr][w];
    uint32_t l = (w > 0)  ? bufA[lr][w - 1] : 0u;
    uint32_t r = (w < 31) ? bufA[lr][w + 1] : 0u;
    bufB[lr][w] = hdil5(m, l, r);
  }
  __syncthreads();

  // ---- Stage 2b: vertical dilation : bufB(hdil) -> bufA(dil) ----
  // dil row dr corresponds to global row (row0 - 2 + dr).
  for (int i = tid; i < DIL_ROWS * WORDS; i += NTHREADS) {
    int dr = i / WORDS, w = i % WORDS;
    bufA[dr][w] = bufB[dr][w] | bufB[dr + 1][w] | bufB[dr + 2][w]
                | bufB[dr + 3][w] | bufB[dr + 4][w];
  }
  __syncthreads();

  // ---- Stage 3: horizontal erosion (pad = foreground) : bufA(dil) -> bufB(hero)
  for (int i = tid; i < DIL_ROWS * WORDS; i += NTHREADS) {
    int dr = i / WORDS, w = i % WORDS;
    int g = row0 - 2 + dr;
    uint32_t h;
    if (g < 0 || g >= HDIM) {
      h = 0xFFFFFFFFu;                       // rows outside image: all foreground
    } else {
      uint32_t nd = ~bufA[dr][w];
      uint32_t nl = (w > 0)  ? ~bufA[dr][w - 1] : 0u;   // col pad: ~x = 0 -> x = 1
      uint32_t nr = (w < 31) ? ~bufA[dr][w + 1] : 0u;
      h = ~hdil5(nd, nl, nr);                // erosion = ~dilate(~x)
    }
    bufB[dr][w] = h;
  }
  __syncthreads();

  // ---- Stage 4: vertical erosion + write ch0 = closed ? 0 : 1 (float4) ----
  float* dst = out + (size_t)img * IMG_STRIDE;  // channel 0 of image
  for (int it = 0; it < (TILE_ROWS * WDIM / 4) / NTHREADS; ++it) {
    int q   = it * NTHREADS + tid;           // float4 index within band
    int p   = q * 4;
    int ro  = p / WDIM;                      // output row in band
    int col = p % WDIM;
    int w   = col >> 5;
    // output row ro -> dil row ro+2 -> hero rows ro..ro+4
    uint32_t closed = bufB[ro][w] & bufB[ro + 1][w] & bufB[ro + 2][w]
                    & bufB[ro + 3][w] & bufB[ro + 4][w];
    uint32_t bits = closed >> (col & 31);
    float4 o;
    o.x = (bits & 1u) ? 0.0f : 1.0f;
    o.y = (bits & 2u) ? 0.0f : 1.0f;
    o.z = (bits & 4u) ? 0.0f : 1.0f;
    o.w = (bits & 8u) ? 0.0f : 1.0f;
    *(float4*)(dst + (size_t)(row0 + ro) * WDIM + col) = o;
  }
}

// ---- Channel 1 pass-through: pure streaming copy ----
__global__ __launch_bounds__(NTHREADS)
void BridgeGap_copy_ch1_kernel(const float* __restrict__ in,
                               float* __restrict__ out) {
  size_t f4      = (size_t)blockIdx.x * NTHREADS + threadIdx.x;
  size_t per_img = (size_t)CH_STRIDE / 4;            // 262144 float4 per image ch
  size_t img     = f4 / per_img;
  size_t rem     = f4 - img * per_img;
  size_t off     = img * (size_t)IMG_STRIDE + CH_STRIDE + rem * 4;
  *(float4*)(out + off) = *(const float4*)(in + off);
}

extern "C" void kernel_launch(void* const* d_in, const int* in_sizes, int n_in,
                              void* d_out, int out_size, void* d_ws, size_t ws_size,
                              hipStream_t stream) {
  (void)in_sizes; (void)n_in; (void)out_size; (void)d_ws; (void)ws_size;
  const float* in = (const float*)d_in[0];
  float* out = (float*)d_out;

  // ch1 copy: 16 * 1024 * 1024 / 4 float4 = 4,194,304 -> 16384 blocks
  BridgeGap_copy_ch1_kernel<<<(BATCH * CH_STRIDE / 4) / NTHREADS, NTHREADS, 0, stream>>>(in, out);
  // ch0 morphology: 16 images * 32 bands = 512 blocks
  BridgeGap_close_ch0_kernel<<<BATCH * (HDIM / TILE_ROWS), NTHREADS, 0, stream>>>(in, out);
}